// TripleConnect_73340861546847
// MI455X (gfx1250) — compile-verified
//
#include <hip/hip_runtime.h>
#include <math.h>
#include <stdint.h>

// Problem constants (from reference): B=8, N=8192, S=4, D=256
#define BB 8
#define NN 8192
#define SS 4
#define DD 256
#define TD 768                  // 3*D
#define MT 32                   // rows per workgroup (2 WMMA M-tiles)
#define LDS1_STRIDE 772         // 768 + 4 pad -> conflict-free ds_load_b64 A fragments
#define LDS2_STRIDE 260         // 256 + 4 pad

typedef float v2f __attribute__((ext_vector_type(2)));
typedef float v8f __attribute__((ext_vector_type(8)));

#define WMMA_F32(A, Bf, C) \
    __builtin_amdgcn_wmma_f32_16x16x4_f32(false, (A), false, (Bf), (short)0, (C), false, false)

// Re-pack a [K x 256] row-major weight matrix into WMMA-B-fragment order:
// Wp[ntile][kstep][lane*2+j] = W[(kstep*4 + 2*(lane>>4) + j)*256 + ntile*16 + (lane&15)]
// so the per-K-step B operand is one contiguous global_load_b64 per lane.
__global__ __launch_bounds__(256) void pack_w_kernel(const float* __restrict__ W,
                                                     float* __restrict__ Wp, int K) {
    int idx = blockIdx.x * blockDim.x + threadIdx.x;   // [0, 16 * K/4 * 64)
    int ksteps = K >> 2;
    int lane2 = idx & 63;
    int lane  = lane2 >> 1;
    int j     = lane2 & 1;
    int t     = idx >> 6;
    int kstep = t % ksteps;
    int ntile = t / ksteps;
    int k = (kstep << 2) + ((lane >> 4) << 1) + j;
    int n = (ntile << 4) + (lane & 15);
    Wp[idx] = W[k * DD + n];
}

__global__ __launch_bounds__(256) void triple_connect_kernel(
        const float* __restrict__ x,
        const int*   __restrict__ ji,
        const int*   __restrict__ ki,
        const float* __restrict__ b1,
        const float* __restrict__ b2,
        const float* __restrict__ W1p,
        const float* __restrict__ W2p,
        float* __restrict__ out) {
    __shared__ __align__(16) float ldsA[MT * LDS1_STRIDE];  // 32 x 768 gathered h tile
    __shared__ __align__(16) float ldsH[MT * LDS2_STRIDE];  // 32 x 256 GELU activations

    const int tid  = threadIdx.x;
    const int lane = tid & 31;
    const int wave = tid >> 5;
    const int half = lane >> 4;        // 0: lanes 0-15, 1: lanes 16-31
    const int col  = lane & 15;
    const int tile = blockIdx.x;       // 32 rows of the (B*N*S) row space

    // ------------- Gather via CDNA5 async-to-LDS: h tile = [xi | xj | xk] ---------
    {
        const int r     = tid >> 3;          // row 0..31 within tile (8 threads/row)
        const int cbase = tid & 7;           // float4 column phase
        const int g = (tile << 5) + r;       // global row index in [0, B*N*S)
        const int b = g >> 15;               // / (N*S)
        const int n = (g >> 2) & (NN - 1);
        const long ibase = ((long)b * NN + n) * SS + (g & 3);
        const int jrow = ji[ibase];
        const int krow = ki[ibase];
        const float* xbase = x + ((long)b * NN) * DD;
        const float* seg_ptr[3] = { xbase + (long)n * DD,
                                    xbase + (long)jrow * DD,
                                    xbase + (long)krow * DD };
        #pragma unroll
        for (int ii = 0; ii < 24; ii++) {
            int c4  = cbase + (ii << 3);     // 0..191 float4 within row
            int seg = c4 >> 6;               // 0=xi, 1=xj, 2=xk (compile-time per ii)
            int c   = c4 & 63;
            unsigned long long src =
                (unsigned long long)(uintptr_t)(const void*)(seg_ptr[seg] + (c << 2));
            unsigned dst =
                (unsigned)(uintptr_t)(const void*)&ldsA[r * LDS1_STRIDE + (c4 << 2)];
            asm volatile("global_load_async_to_lds_b128 %0, %1, off"
                         :: "v"(dst), "v"(src) : "memory");
        }
        asm volatile("s_wait_asynccnt 0x0" ::: "memory");
    }
    __syncthreads();

    // -------- GEMM1: (32x768) @ (768x256), per-wave 2 M-tiles x 2 N-tiles ---------
    const int nt0 = wave << 1;
    v8f a00, a01, a10, a11;   // [mtile][ntile]
    {
        float bv0 = b1[(nt0 << 4) + col];
        float bv1 = b1[((nt0 + 1) << 4) + col];
        #pragma unroll
        for (int v = 0; v < 8; v++) { a00[v] = bv0; a01[v] = bv1; a10[v] = bv0; a11[v] = bv1; }
    }
    {
        const float* arow0 = ldsA + col * LDS1_STRIDE + (half << 1);
        const float* arow1 = ldsA + (col + 16) * LDS1_STRIDE + (half << 1);
        const float* bp0   = W1p + ((nt0 * 192) << 6) + (lane << 1);
        const float* bp1   = W1p + (((nt0 + 1) * 192) << 6) + (lane << 1);
        #pragma unroll 2
        for (int ks = 0; ks < 192; ks++) {
            v2f af0 = *(const v2f*)(arow0 + (ks << 2));
            v2f af1 = *(const v2f*)(arow1 + (ks << 2));
            v2f bf0 = *(const v2f*)(bp0 + (ks << 6));
            v2f bf1 = *(const v2f*)(bp1 + (ks << 6));
            a00 = WMMA_F32(af0, bf0, a00);
            a01 = WMMA_F32(af0, bf1, a01);
            a10 = WMMA_F32(af1, bf0, a10);
            a11 = WMMA_F32(af1, bf1, a11);
        }
    }

    // ---------------- Exact GELU, stage activations through LDS --------------------
    #pragma unroll
    for (int v = 0; v < 8; v++) {
        int row0 = v + (half << 3);          // rows 0..15  (M-tile 0)
        int row1 = row0 + 16;                // rows 16..31 (M-tile 1)
        float t00 = a00[v], t01 = a01[v], t10 = a10[v], t11 = a11[v];
        t00 = 0.5f * t00 * (1.0f + erff(t00 * 0.70710678118654752f));
        t01 = 0.5f * t01 * (1.0f + erff(t01 * 0.70710678118654752f));
        t10 = 0.5f * t10 * (1.0f + erff(t10 * 0.70710678118654752f));
        t11 = 0.5f * t11 * (1.0f + erff(t11 * 0.70710678118654752f));
        ldsH[row0 * LDS2_STRIDE + (nt0 << 4) + col]       = t00;
        ldsH[row0 * LDS2_STRIDE + ((nt0 + 1) << 4) + col] = t01;
        ldsH[row1 * LDS2_STRIDE + (nt0 << 4) + col]       = t10;
        ldsH[row1 * LDS2_STRIDE + ((nt0 + 1) << 4) + col] = t11;
    }
    __syncthreads();

    // ---------------- GEMM2: (32x256) @ (256x256) ---------------------------------
    {
        float bv0 = b2[(nt0 << 4) + col];
        float bv1 = b2[((nt0 + 1) << 4) + col];
        #pragma unroll
        for (int v = 0; v < 8; v++) { a00[v] = bv0; a01[v] = bv1; a10[v] = bv0; a11[v] = bv1; }
    }
    {
        const float* arow0 = ldsH + col * LDS2_STRIDE + (half << 1);
        const float* arow1 = ldsH + (col + 16) * LDS2_STRIDE + (half << 1);
        const float* bp0   = W2p + ((nt0 * 64) << 6) + (lane << 1);
        const float* bp1   = W2p + (((nt0 + 1) * 64) << 6) + (lane << 1);
        #pragma unroll 2
        for (int ks = 0; ks < 64; ks++) {
            v2f af0 = *(const v2f*)(arow0 + (ks << 2));
            v2f af1 = *(const v2f*)(arow1 + (ks << 2));
            v2f bf0 = *(const v2f*)(bp0 + (ks << 6));
            v2f bf1 = *(const v2f*)(bp1 + (ks << 6));
            a00 = WMMA_F32(af0, bf0, a00);
            a01 = WMMA_F32(af0, bf1, a01);
            a10 = WMMA_F32(af1, bf0, a10);
            a11 = WMMA_F32(af1, bf1, a11);
        }
    }

    // ---------------- Mean over S=4 (pure in-register) and store -------------------
    // C fragment: VGPR v of lane holds row (v + 8*half) of its M-tile,
    // col = ntile*16 + (lane&15). Rows 4g..4g+3 are consecutive s for one (b,n).
    {
        float g0a = 0.25f * (a00[0] + a00[1] + a00[2] + a00[3]);
        float g1a = 0.25f * (a00[4] + a00[5] + a00[6] + a00[7]);
        float g0b = 0.25f * (a01[0] + a01[1] + a01[2] + a01[3]);
        float g1b = 0.25f * (a01[4] + a01[5] + a01[6] + a01[7]);
        long orow = (long)((tile << 3) + (half << 1));         // M-tile 0
        long obase = orow * DD;
        out[obase + (nt0 << 4) + col]            = g0a;
        out[obase + DD + (nt0 << 4) + col]       = g1a;
        out[obase + ((nt0 + 1) << 4) + col]      = g0b;
        out[obase + DD + ((nt0 + 1) << 4) + col] = g1b;
    }
    {
        float g0a = 0.25f * (a10[0] + a10[1] + a10[2] + a10[3]);
        float g1a = 0.25f * (a10[4] + a10[5] + a10[6] + a10[7]);
        float g0b = 0.25f * (a11[0] + a11[1] + a11[2] + a11[3]);
        float g1b = 0.25f * (a11[4] + a11[5] + a11[6] + a11[7]);
        long orow = (long)((tile << 3) + 4 + (half << 1));     // M-tile 1
        long obase = orow * DD;
        out[obase + (nt0 << 4) + col]            = g0a;
        out[obase + DD + (nt0 << 4) + col]       = g1a;
        out[obase + ((nt0 + 1) << 4) + col]      = g0b;
        out[obase + DD + ((nt0 + 1) << 4) + col] = g1b;
    }
}

extern "C" void kernel_launch(void* const* d_in, const int* in_sizes, int n_in,
                              void* d_out, int out_size, void* d_ws, size_t ws_size,
                              hipStream_t stream) {
    const float* x  = (const float*)d_in[0];
    const int*   ji = (const int*)d_in[1];
    const int*   ki = (const int*)d_in[2];
    const float* W1 = (const float*)d_in[3];
    const float* b1 = (const float*)d_in[4];
    const float* W2 = (const float*)d_in[5];
    const float* b2 = (const float*)d_in[6];
    float* out = (float*)d_out;

    float* W1p = (float*)d_ws;                                            // 768*256 floats
    float* W2p = (float*)((char*)d_ws + (size_t)TD * DD * sizeof(float)); // 256*256 floats

    pack_w_kernel<<<(TD * DD) / 256, 256, 0, stream>>>(W1, W1p, TD);
    pack_w_kernel<<<(DD * DD) / 256, 256, 0, stream>>>(W2, W2p, DD);

    const int n_tiles = (BB * NN * SS) / MT;                              // 8192 blocks
    triple_connect_kernel<<<n_tiles, 256, 0, stream>>>(x, ji, ki, b1, b2, W1p, W2p, out);
}